// BilateralOTSplit_2259152798351
// MI455X (gfx1250) — compile-verified
//
#include <hip/hip_runtime.h>

#define B_    64
#define C_    2048
#define HW_   288
#define CS_   8            // channel chunks in pass 1
#define CCH_  (C_ / CS_)   // 256 channels per chunk
#define EPS_  0.07f
#define ITERS_ 7
#define TINY_ 1e-8f

typedef __attribute__((ext_vector_type(2))) float v2f;
typedef __attribute__((ext_vector_type(8))) float v8f;

// ws layout (floats):
//   [0, B*CS*3*HW)        partial {s, d0, d1} per (batch, chunk)
//   [PART_N, PART_N+2*B)  inv_den per (batch, k)
#define PART_N_  (B_ * CS_ * 3 * HW_)
#define DEN_OFF_ PART_N_

// ---------------------------------------------------------------------------
// Pass 1: per-token partial reductions over a 256-channel chunk.
// Fully coalesced: lanes span hw (contiguous), loop walks channels.
// ---------------------------------------------------------------------------
__global__ __launch_bounds__(HW_) void ot_pass1(const float* __restrict__ x,
                                                const float* __restrict__ anchors,
                                                float* __restrict__ ws) {
  const int b     = blockIdx.x;
  const int chunk = blockIdx.y;
  const int cbase = chunk * CCH_;
  const int hw    = threadIdx.x;

  __shared__ float a0s[CCH_], a1s[CCH_];
  if (hw < CCH_) {
    a0s[hw] = anchors[cbase + hw];
    a1s[hw] = anchors[C_ + cbase + hw];
  }
  __syncthreads();

  const float* xp = x + ((size_t)b * C_ + cbase) * HW_ + hw;
  float s = 0.f, d0 = 0.f, d1 = 0.f;
#pragma unroll 8
  for (int c = 0; c < CCH_; ++c) {
    const float v = xp[(size_t)c * HW_];
    s  = fmaf(v, v,      s);
    d0 = fmaf(v, a0s[c], d0);
    d1 = fmaf(v, a1s[c], d1);
  }
  float* part = ws + (size_t)(b * CS_ + chunk) * 3 * HW_;
  part[hw]           = s;
  part[HW_ + hw]     = d0;
  part[2 * HW_ + hw] = d1;
}

// ---------------------------------------------------------------------------
// Deterministic 288-thread block reduction of two floats (fixed order).
// ---------------------------------------------------------------------------
__device__ __forceinline__ void block_reduce2(float& x0, float& x1, float* sh) {
#pragma unroll
  for (int off = 16; off > 0; off >>= 1) {
    x0 += __shfl_down(x0, off, 32);
    x1 += __shfl_down(x1, off, 32);
  }
  const int warp = threadIdx.x >> 5;
  const int lane = threadIdx.x & 31;
  if (lane == 0) { sh[warp] = x0; sh[16 + warp] = x1; }
  __syncthreads();
  if (threadIdx.x == 0) {
    float t0 = 0.f, t1 = 0.f;
    for (int w = 0; w < (HW_ / 32); ++w) { t0 += sh[w]; t1 += sh[16 + w]; }
    sh[32] = t0; sh[33] = t1;
  }
  __syncthreads();
  x0 = sh[32]; x1 = sh[33];
  __syncthreads();
}

// ---------------------------------------------------------------------------
// Pass 2: assemble dots, cosine cost, 7 Sinkhorn iterations, emit m and
// 1/(den+1e-6). One workgroup per batch; thread == token.
// ---------------------------------------------------------------------------
__global__ __launch_bounds__(HW_) void ot_sinkhorn(const float* __restrict__ anchors,
                                                   float* __restrict__ ws,
                                                   float* __restrict__ out) {
  __shared__ float sh[34];
  const int b  = blockIdx.x;
  const int hw = threadIdx.x;

  // fixed-order sum of the 8 channel-chunk partials
  float s = 0.f, d0 = 0.f, d1 = 0.f;
  for (int chunk = 0; chunk < CS_; ++chunk) {
    const float* part = ws + (size_t)(b * CS_ + chunk) * 3 * HW_;
    s  += part[hw];
    d0 += part[HW_ + hw];
    d1 += part[2 * HW_ + hw];
  }

  // anchor norms (redundant per block, trivial cost)
  float pa = 0.f, pb = 0.f;
  for (int c = hw; c < C_; c += HW_) {
    const float va = anchors[c];      pa = fmaf(va, va, pa);
    const float vb = anchors[C_ + c]; pb = fmaf(vb, vb, pb);
  }
  block_reduce2(pa, pb, sh);
  const float na0 = fmaxf(sqrtf(pa), 1e-12f);
  const float na1 = fmaxf(sqrtf(pb), 1e-12f);
  const float xn  = fmaxf(sqrtf(s), 1e-12f);

  const float cos0 = d0 / (xn * na0);
  const float cos1 = d1 / (xn * na1);
  const float k0 = expf(-(1.f - cos0) / EPS_) + TINY_;
  const float k1 = expf(-(1.f - cos1) / EPS_) + TINY_;

  const float a = 1.0f / (float)HW_;
  float v0 = 1.f, v1 = 1.f, u = 1.f;
  for (int it = 0; it < ITERS_; ++it) {
    const float Kv = fmaf(k0, v0, k1 * v1) + TINY_;
    u = a / Kv;
    float t0 = k0 * u, t1 = k1 * u;
    block_reduce2(t0, t1, sh);
    v0 = 0.5f / (t0 + TINY_);
    v1 = 0.5f / (t1 + TINY_);
  }

  const float m0 = k0 * u * v0 * (float)HW_;
  const float m1 = k1 * u * v1 * (float)HW_;
  float r0 = m0, r1 = m1;
  block_reduce2(r0, r1, sh);
  if (hw == 0) {
    ws[DEN_OFF_ + 2 * b]     = 1.0f / (r0 + 1e-6f);
    ws[DEN_OFF_ + 2 * b + 1] = 1.0f / (r1 + 1e-6f);
  }

  float* mOut = out + 2 * (size_t)B_ * C_ + (size_t)b * 2 * HW_;
  mOut[hw]       = m0;   // m[b,0,hw]
  mOut[HW_ + hw] = m1;   // m[b,1,hw]
}

// ---------------------------------------------------------------------------
// Pass 3: masked pooling as per-batch GEMM with v_wmma_f32_16x16x4_f32.
//   D(16ch x 16) += A(16ch x 4hw) * B(4hw x 16), B cols 0/1 = m rows, rest 0.
// One wave per 16-channel tile; 72 K-steps cover HW=288. x is L2-resident
// from pass 1 (151 MB < 192 MB L2).
// ---------------------------------------------------------------------------
__global__ __launch_bounds__(256) void ot_pool_wmma(const float* __restrict__ x,
                                                    const float* __restrict__ ws,
                                                    float* __restrict__ out) {
  __shared__ __align__(16) float mtile[2 * HW_];
  __shared__ __align__(16) float zrow[HW_];
  __shared__ float inv2[2];

  const int b   = blockIdx.x >> 4;   // 16 blocks per batch
  const int g   = blockIdx.x & 15;
  const int tid = threadIdx.x;

  const float* mGlob = out + 2 * (size_t)B_ * C_ + (size_t)b * 2 * HW_;
  for (int i = tid; i < 2 * HW_; i += 256) mtile[i] = mGlob[i];
  for (int i = tid; i < HW_;     i += 256) zrow[i]  = 0.f;
  if (tid < 2) inv2[tid] = ws[DEN_OFF_ + 2 * b + tid];
  __syncthreads();

  const int wave = tid >> 5;
  const int lane = tid & 31;
  const int n    = lane & 15;   // A: M row within half / B,D: column N
  const int hi   = lane >> 4;   // K half select (A/B), M+8 select (D)
  const int cbase = (g * 8 + wave) * 16;   // 128 channel tiles per batch

  // A lane source: x[b, cbase+n, hw0 + 2*hi + {0,1}]
  const float* xrow = x + ((size_t)b * C_ + (cbase + n)) * HW_ + 2 * hi;
  // B lane source: columns 0/1 carry m[b, n, :]; columns 2..15 read zeros
  const float* brow = (n < 2) ? (mtile + n * HW_ + 2 * hi) : (zrow + 2 * hi);

  v8f acc = {};
#pragma unroll 4
  for (int hw0 = 0; hw0 < HW_; hw0 += 4) {
    const float2 xv = *(const float2*)(xrow + hw0);
    const float2 bv = *(const float2*)(brow + hw0);
    v2f av; av.x = xv.x; av.y = xv.y;
    v2f bb; bb.x = bv.x; bb.y = bv.y;
    acc = __builtin_amdgcn_wmma_f32_16x16x4_f32(
        /*neg_a=*/false, av, /*neg_b=*/false, bb,
        /*c_mod=*/(short)0, acc, /*reuse_a=*/false, /*reuse_b=*/false);
  }

  // D layout: VGPR r on lanes (n, n+16) holds rows M=r and M=8+r of column n.
  // Columns 0/1 are the two prototypes -> 8 contiguous floats per lane.
  if (n < 2) {
    const float scale = inv2[n];
    float* op = out + (size_t)n * B_ * C_ + (size_t)b * C_ + cbase + hi * 8;
#pragma unroll
    for (int r = 0; r < 8; ++r) op[r] = acc[r] * scale;
  }
}

// ---------------------------------------------------------------------------
extern "C" void kernel_launch(void* const* d_in, const int* in_sizes, int n_in,
                              void* d_out, int out_size, void* d_ws, size_t ws_size,
                              hipStream_t stream) {
  (void)in_sizes; (void)n_in; (void)out_size; (void)ws_size;
  const float* x       = (const float*)d_in[0];
  const float* anchors = (const float*)d_in[1];
  float* out = (float*)d_out;
  float* ws  = (float*)d_ws;

  ot_pass1<<<dim3(B_, CS_), HW_, 0, stream>>>(x, anchors, ws);
  ot_sinkhorn<<<B_, HW_, 0, stream>>>(anchors, ws, out);
  ot_pool_wmma<<<B_ * 16, 256, 0, stream>>>(x, ws, out);
}